// GatedRMSNorm2d_2877628088595
// MI455X (gfx1250) — compile-verified
//
#include <hip/hip_runtime.h>

// Problem dims (fixed by reference)
#define B_  8
#define C_  256
#define H_  128
#define W_  128
#define R_  16
#define HW_ (H_*W_)

// LDS layout (floats), padded strides to avoid bank conflicts
#define YSTRIDE   132          // 128 px + pad (132*4 B = 16B-aligned rows)
#define WD_STRIDE 257          // w_down row stride (16 rows)
#define WU_STRIDE 18           // w_up row stride (256 rows)
#define SB_PXS    17           // silu scratch px stride
#define SB_WSTRIDE (R_*SB_PXS) // per-wave silu scratch

#define OFF_Y    0
#define OFF_SS   (OFF_Y   + C_*YSTRIDE)     // 32 sub x 8 grp x 4 px partials
#define OFF_RINV (OFF_SS  + 1024)           // 128 rsqrt values
#define OFF_NW   (OFF_RINV+ 128)            // 256 norm weights
#define OFF_BU   (OFF_NW  + 256)            // 256 bias
#define OFF_WD   (OFF_BU  + 256)            // 16x257
#define OFF_WU   (OFF_WD  + R_*WD_STRIDE)   // 256x18
#define OFF_SB   (OFF_WU  + C_*WU_STRIDE)   // 8 waves x 16x17
#define SMEM_FLOATS (OFF_SB + 8*SB_WSTRIDE)

typedef float v2f __attribute__((ext_vector_type(2)));
typedef float v4f __attribute__((ext_vector_type(4)));
typedef float v8f __attribute__((ext_vector_type(8)));

// Fast sigmoid: v_exp_f32 + v_rcp_f32 (no IEEE division sequence)
__device__ __forceinline__ float fast_sigmoid(float v) {
    return __builtin_amdgcn_rcpf(1.0f + __expf(-v));
}

__global__ __launch_bounds__(256)
void gated_rmsnorm2d_kernel(const float* __restrict__ x,
                            const float* __restrict__ nw,
                            const float* __restrict__ wd,
                            const float* __restrict__ wu,
                            const float* __restrict__ bu,
                            float* __restrict__ out)
{
    extern __shared__ float sm[];
    const int t   = threadIdx.x;
    const int blk = blockIdx.x;        // b*H + h
    const int b   = blk >> 7;
    const int h   = blk & (H_-1);
    const long xbase = (long)b * (C_*HW_) + (long)h * W_;

    // ---- Phase 0: preload weights into LDS ----
    sm[OFF_NW + t] = nw[t];
    sm[OFF_BU + t] = bu[t];
#pragma unroll
    for (int i = 0; i < 16; ++i) {
        int idx = t + i*256;
        sm[OFF_WD + (idx >> 8)*WD_STRIDE + (idx & 255)] = wd[idx]; // [r=16][c=256]
        sm[OFF_WU + (idx >> 4)*WU_STRIDE + (idx & 15)]  = wu[idx]; // [c=256][r=16]
    }

    // ---- Phase 1: b128 tile load + per-pixel sum of squares ----
    // thread t: pixel quad px = sub*4..sub*4+3, channels c = 8*i + grp
    const int sub = t & 31;
    const int grp = t >> 5;
    v4f ss4 = {};
    for (int i = 0; i < 32; ++i) {
        int c = 8*i + grp;
        v4f v = *(const v4f*)(x + xbase + (long)c*HW_ + sub*4);
        *(v4f*)(&sm[OFF_Y + c*YSTRIDE + sub*4]) = v;
        ss4 += v*v;
    }
    *(v4f*)(&sm[OFF_SS + (sub*8 + grp)*4]) = ss4;
    __syncthreads();
    if (t < 128) {
        int psub = t >> 2, pj = t & 3;
        float tot = 0.f;
#pragma unroll
        for (int g = 0; g < 8; ++g)
            tot += sm[OFF_SS + (psub*8 + g)*4 + pj];
        sm[OFF_RINV + t] = rsqrtf(tot * (1.0f/(float)C_) + 1e-6f);
    }
    __syncthreads();

    // ---- Phase 2: normalize in place (b128 LDS RMW): y = x * rinv * nw[c] ----
    {
        v4f rv4 = *(const v4f*)(&sm[OFF_RINV + sub*4]);
        for (int i = 0; i < 32; ++i) {
            int c = 8*i + grp;
            v4f* p = (v4f*)(&sm[OFF_Y + c*YSTRIDE + sub*4]);
            *p = *p * (rv4 * sm[OFF_NW + c]);
        }
    }
    __syncthreads();

    // ---- Phase 3: per-wave down-projection via f32 WMMA ----
    // Wave j owns pixels [j*16, j*16+16). D[16r x 16px] = Wd[16x256] * Y[256x16]
    const int wave  = t >> 5;
    const int lane  = t & 31;
    const int lhalf = lane >> 4;     // 0: K=+0,+1   1: K=+2,+3  (32-bit A layout)
    const int lidx  = lane & 15;     // row (A) / col (B,C,D)
    const int px0   = wave * 16;
    const int px    = px0 + lidx;

    v8f dacc = {};
    const int arow = OFF_WD + lidx*WD_STRIDE;
#pragma unroll 4
    for (int kc = 0; kc < C_; kc += 4) {
        int ka = kc + lhalf*2;
        v2f a, bb;
        a.x  = sm[arow + ka];
        a.y  = sm[arow + ka + 1];
        bb.x = sm[OFF_Y + ka*YSTRIDE + px];
        bb.y = sm[OFF_Y + (ka+1)*YSTRIDE + px];
        dacc = __builtin_amdgcn_wmma_f32_16x16x4_f32(
                   false, a, false, bb, (short)0, dacc, false, false);
    }

    // SiLU on D, stash S[16x16] in per-wave LDS scratch (D layout -> row major)
    const int sbase = OFF_SB + wave*SB_WSTRIDE;
#pragma unroll
    for (int i = 0; i < 8; ++i) {
        float dv = dacc[i];
        float s  = dv * fast_sigmoid(dv);           // silu via v_exp + v_rcp
        sm[sbase + (lhalf*8 + i)*SB_PXS + lidx] = s;
    }

    // ---- Phase 4: up-projection (16 M-tiles of 16 channels), gate,
    //      write g*y back in place into the y-tile (each (c,px) owned by one lane)
#pragma unroll 1
    for (int ci = 0; ci < 16; ++ci) {
        v8f uacc = {};
        const int ca = (ci*16 + lidx)*WU_STRIDE + OFF_WU;
#pragma unroll
        for (int rc = 0; rc < R_; rc += 4) {
            int kk = rc + lhalf*2;
            v2f a, bb;
            a.x  = sm[ca + kk];
            a.y  = sm[ca + kk + 1];
            bb.x = sm[sbase + kk*SB_PXS + lidx];
            bb.y = sm[sbase + (kk+1)*SB_PXS + lidx];
            uacc = __builtin_amdgcn_wmma_f32_16x16x4_f32(
                       false, a, false, bb, (short)0, uacc, false, false);
        }
#pragma unroll
        for (int i = 0; i < 8; ++i) {
            int   c  = ci*16 + lhalf*8 + i;
            float u  = uacc[i] + sm[OFF_BU + c];
            float g  = fast_sigmoid(u);             // sigmoid via v_exp + v_rcp
            float yv = sm[OFF_Y + c*YSTRIDE + px];
            sm[OFF_Y + c*YSTRIDE + px] = g * yv;    // in-place result
        }
    }
    __syncthreads();

    // ---- Phase 5: cooperative b128 store of the result tile ----
    for (int i = 0; i < 32; ++i) {
        int c = 8*i + grp;
        *(v4f*)(out + xbase + (long)c*HW_ + sub*4) =
            *(const v4f*)(&sm[OFF_Y + c*YSTRIDE + sub*4]);
    }
}

extern "C" void kernel_launch(void* const* d_in, const int* in_sizes, int n_in,
                              void* d_out, int out_size, void* d_ws, size_t ws_size,
                              hipStream_t stream) {
    (void)in_sizes; (void)n_in; (void)out_size; (void)d_ws; (void)ws_size;
    const float* x  = (const float*)d_in[0];
    const float* nw = (const float*)d_in[1];
    const float* wd = (const float*)d_in[2];
    const float* wu = (const float*)d_in[3];
    const float* bu = (const float*)d_in[4];
    float* out = (float*)d_out;

    const size_t smem_bytes = (size_t)SMEM_FLOATS * sizeof(float); // ~185 KB < 320 KB/WGP
    hipFuncSetAttribute(reinterpret_cast<const void*>(gated_rmsnorm2d_kernel),
                        hipFuncAttributeMaxDynamicSharedMemorySize, (int)smem_bytes);

    dim3 grid(B_ * H_);   // 1024 workgroups, one (b,h) row each
    dim3 block(256);      // 8 wave32
    gated_rmsnorm2d_kernel<<<grid, block, smem_bytes, stream>>>(x, nw, wd, wu, bu, out);
}